// GGCN_48679159333162
// MI455X (gfx1250) — compile-verified
//
#include <hip/hip_runtime.h>
#include <math.h>

typedef __attribute__((ext_vector_type(2))) float v2f;
typedef __attribute__((ext_vector_type(8))) float v8f;

constexpr int CN   = 128;   // channels
constexpr int TT   = 16;    // node types
constexpr int BB   = 256;   // graphs
constexpr int SEGS = BB * TT;

// ---------------- 1) node embedding: x = [type_emb | mean(attr_emb)] ----------------
__global__ void k_embed(const int* __restrict__ x_type, const int* __restrict__ x_attrs,
                        const int* __restrict__ x_attrs_mask,
                        const float* __restrict__ type_emb, const float* __restrict__ attr_emb,
                        float* __restrict__ x, int N) {
    int n = blockIdx.x;
    int c = threadIdx.x;            // 0..127
    if (n >= N) return;
    float v;
    if (c < 64) {
        v = type_emb[x_type[n] * 64 + c];
    } else {
        int mask = x_attrs_mask[n]; // 1..8
        float s = 0.f;
        for (int a = 0; a < mask; ++a)
            s += attr_emb[x_attrs[n * 8 + a] * 64 + (c - 64)];
        v = s / (float)mask;
    }
    x[n * CN + c] = v;
}

// ---------------- 2) message GEMM: H = X @ W  (W row-major [128][128]) --------------
// block = 256 threads = 8 waves; block computes rows [16) x cols [128); wave -> 16x16 tile
__global__ void k_msg_gemm(const float* __restrict__ X, const float* __restrict__ W,
                           float* __restrict__ H, int N) {
    __shared__ float lx[16 * CN];
    int row0 = blockIdx.x * 16;
    int tid  = threadIdx.x;
    for (int i = tid; i < 16 * CN; i += 256) {
        int r = i >> 7, cc = i & 127;
        int gr = row0 + r;
        lx[i] = (gr < N) ? X[gr * CN + cc] : 0.f;
    }
    __syncthreads();
    int wave = tid >> 5, lane = tid & 31;
    int col0 = wave * 16;
    int ml   = lane & 15;
    int khi  = (lane >> 4) * 2;     // 0 or 2
    v8f acc = {};
    for (int k = 0; k < CN; k += 4) {
        v2f a, b;
        a.x = lx[ml * CN + k + khi];
        a.y = lx[ml * CN + k + khi + 1];
        b.x = W[(k + khi)     * CN + col0 + ml];
        b.y = W[(k + khi + 1) * CN + col0 + ml];
        acc = __builtin_amdgcn_wmma_f32_16x16x4_f32(false, a, false, b, (short)0, acc, false, false);
    }
    int rbase = row0 + ((lane >> 4) << 3);
#pragma unroll
    for (int i = 0; i < 8; ++i) {
        int r = rbase + i;
        if (r < N) H[r * CN + col0 + ml] = acc[i];
    }
}

// ---------------- 3) edge scatter for one edge type: m[dst] += H[src] ---------------
// one wave per edge; lane handles 4 channels (float4 gather + 4 f32 atomics)
__global__ void k_scatter(const float* __restrict__ H, const int* __restrict__ src,
                          const int* __restrict__ dst, const int* __restrict__ etype,
                          float* __restrict__ m, int E, int t) {
    int e = blockIdx.x * 8 + (threadIdx.x >> 5);
    if (e >= E) return;
    if (etype[e] != t) return;      // wave-uniform branch
    int lane = threadIdx.x & 31;
    int s = src[e], d = dst[e];
    const float4* hp = (const float4*)(H + s * CN);
    float4 v = hp[lane];
    float* mp = m + d * CN + lane * 4;
    atomicAdd(mp + 0, v.x);
    atomicAdd(mp + 1, v.y);
    atomicAdd(mp + 2, v.z);
    atomicAdd(mp + 3, v.w);
}

// ---------------- 4) fused GRU cell: x = GRU(m, x) (in place) -----------------------
// block = 256 = 8 waves per 16-node tile. waves 0..3: gi = m@Wih^T ; waves 4..7: gh = x@Whh^T
__global__ void k_gru(const float* __restrict__ M, float* __restrict__ Xio,
                      const float* __restrict__ Wih, const float* __restrict__ Whh,
                      const float* __restrict__ bih, const float* __restrict__ bhh,
                      int N) {
    __shared__ float lm[16 * CN];
    __shared__ float lh[16 * CN];
    __shared__ float lgi[16 * 384];
    __shared__ float lgh[16 * 384];
    int row0 = blockIdx.x * 16;
    int tid  = threadIdx.x;
    for (int i = tid; i < 16 * CN; i += 256) {
        int r = i >> 7, cc = i & 127;
        int gr = row0 + r;
        lm[i] = (gr < N) ? M[gr * CN + cc] : 0.f;
        lh[i] = (gr < N) ? Xio[gr * CN + cc] : 0.f;
    }
    __syncthreads();
    int wave = tid >> 5, lane = tid & 31;
    int ml  = lane & 15;
    int khi = (lane >> 4) * 2;
    const float* A = (wave < 4) ? (const float*)lm : (const float*)lh;
    const float* B = (wave < 4) ? Wih : Whh;
    float*       G = (wave < 4) ? lgi : lgh;
    int t0 = ((wave < 4) ? wave : (wave - 4)) * 6;   // 6 col-tiles per wave -> 24 tiles = 384 cols
    for (int q = 0; q < 6; ++q) {
        int col0 = (t0 + q) * 16;
        v8f acc = {};
        for (int k = 0; k < CN; k += 4) {
            v2f a, b;
            a.x = A[ml * CN + k + khi];
            a.y = A[ml * CN + k + khi + 1];
            int jc = col0 + ml;                       // B[K=c][col=j] = W[j*128 + c]
            b.x = B[jc * CN + k + khi];
            b.y = B[jc * CN + k + khi + 1];
            acc = __builtin_amdgcn_wmma_f32_16x16x4_f32(false, a, false, b, (short)0, acc, false, false);
        }
        int rb = (lane >> 4) << 3;
#pragma unroll
        for (int i = 0; i < 8; ++i)
            G[(rb + i) * 384 + col0 + ml] = acc[i];
    }
    __syncthreads();
    for (int i = tid; i < 16 * CN; i += 256) {
        int r = i >> 7, c = i & 127;
        int gr = row0 + r;
        if (gr >= N) continue;
        float ir = lgi[r * 384 + c]       + bih[c];
        float iz = lgi[r * 384 + 128 + c] + bih[128 + c];
        float in = lgi[r * 384 + 256 + c] + bih[256 + c];
        float hr = lgh[r * 384 + c]       + bhh[c];
        float hz = lgh[r * 384 + 128 + c] + bhh[128 + c];
        float hn = lgh[r * 384 + 256 + c] + bhh[256 + c];
        float rg = 1.f / (1.f + __expf(-(ir + hr)));
        float zg = 1.f / (1.f + __expf(-(iz + hz)));
        float ng = tanhf(in + rg * hn);
        float h  = lh[i];
        Xio[gr * CN + c] = (1.f - zg) * ng + zg * h;
    }
}

// ---------------- 5) per-(graph,type) segment sums ----------------------------------
__global__ void k_segsum(const float* __restrict__ X, const int* __restrict__ batch,
                         const int* __restrict__ x_type,
                         float* __restrict__ sums, float* __restrict__ cnt, int N) {
    int n = blockIdx.x;
    int c = threadIdx.x;
    if (n >= N) return;
    int seg = batch[n] * TT + x_type[n];
    atomicAdd(&sums[seg * CN + c], X[n * CN + c]);
    if (c == 0) atomicAdd(&cnt[seg], 1.f);
}

// ---------------- 6) mean + attention logits ----------------------------------------
__global__ void k_agg(float* __restrict__ sums, const float* __restrict__ cnt,
                      const float* __restrict__ w_attn, const float* __restrict__ b_attn,
                      float* __restrict__ wlog) {
    __shared__ float red[CN];
    int s = blockIdx.x;             // 0..4095
    int c = threadIdx.x;
    float ct = cnt[s];
    float v  = sums[s * CN + c];
    v = (ct > 0.f) ? v / ct : 0.f;
    sums[s * CN + c] = v;           // agg in place
    red[c] = v * w_attn[c];
    __syncthreads();
    for (int off = 64; off > 0; off >>= 1) {
        if (c < off) red[c] += red[c + off];
        __syncthreads();
    }
    if (c == 0) wlog[s] = red[0] + b_attn[0];
}

// ---------------- 7) softmax + encoder + decoder per graph --------------------------
__global__ void k_head(const float* __restrict__ agg, const float* __restrict__ wlog,
                       const float* __restrict__ W_enc, const float* __restrict__ b_enc,
                       const float* __restrict__ W_dec, const float* __restrict__ b_dec,
                       float* __restrict__ out) {
    __shared__ float attn[TT];
    __shared__ float flat[TT * CN];
    __shared__ float enc[CN];
    int g = blockIdx.x;
    int tid = threadIdx.x;          // 128
    if (tid == 0) {
        float mx = -1e30f;
        for (int t = 0; t < TT; ++t) mx = fmaxf(mx, wlog[g * TT + t]);
        float s = 0.f;
        for (int t = 0; t < TT; ++t) { float e = __expf(wlog[g * TT + t] - mx); attn[t] = e; s += e; }
        for (int t = 0; t < TT; ++t) attn[t] /= s;
    }
    __syncthreads();
    for (int i = tid; i < TT * CN; i += 128) {
        int t = i >> 7, c = i & 127;
        flat[i] = attn[t] * agg[(g * TT + t) * CN + c];
    }
    __syncthreads();
    float acc = b_enc[tid];
    for (int i = 0; i < TT * CN; ++i)
        acc += flat[i] * W_enc[i * CN + tid];
    enc[tid] = acc;
    __syncthreads();
    if (tid < 10) {
        float o = b_dec[tid];
        for (int j = 0; j < CN; ++j)
            o += enc[j] * W_dec[j * 10 + tid];
        out[g * 10 + tid] = o;
    }
}

extern "C" void kernel_launch(void* const* d_in, const int* in_sizes, int n_in,
                              void* d_out, int out_size, void* d_ws, size_t ws_size,
                              hipStream_t stream) {
    const int* x_type       = (const int*)d_in[0];
    const int* x_attrs      = (const int*)d_in[1];
    const int* x_attrs_mask = (const int*)d_in[2];
    const int* edge_index   = (const int*)d_in[3];
    const int* edge_attr    = (const int*)d_in[4];
    const int* batch        = (const int*)d_in[6];
    const float* type_emb   = (const float*)d_in[7];
    const float* attr_emb   = (const float*)d_in[8];
    const float* W_msg      = (const float*)d_in[9];
    const float* W_ih       = (const float*)d_in[10];
    const float* W_hh       = (const float*)d_in[11];
    const float* b_ih       = (const float*)d_in[12];
    const float* b_hh       = (const float*)d_in[13];
    const float* w_attn     = (const float*)d_in[14];
    const float* b_attn     = (const float*)d_in[15];
    const float* W_enc      = (const float*)d_in[16];
    const float* b_enc      = (const float*)d_in[17];
    const float* W_dec      = (const float*)d_in[18];
    const float* b_dec      = (const float*)d_in[19];
    float* out = (float*)d_out;

    const int N = in_sizes[0];
    const int E = in_sizes[4];
    const int L = 4, TE = 4;
    const int* src = edge_index;
    const int* dst = edge_index + E;

    // workspace layout (floats)
    float* ws   = (float*)d_ws;
    float* x    = ws;                       // N*128
    float* H    = x + (size_t)N * CN;       // N*128
    float* m    = H + (size_t)N * CN;       // N*128
    float* sums = m + (size_t)N * CN;       // 4096*128
    float* cnt  = sums + (size_t)SEGS * CN; // 4096
    float* wlog = cnt + SEGS;               // 4096

    int ntiles = (N + 15) / 16;

    k_embed<<<N, CN, 0, stream>>>(x_type, x_attrs, x_attrs_mask, type_emb, attr_emb, x, N);

    for (int l = 0; l < L; ++l) {
        hipMemsetAsync(m, 0, (size_t)N * CN * sizeof(float), stream);
        for (int t = 0; t < TE; ++t) {
            const float* Wt = W_msg + (size_t)(l * TE + t) * CN * CN;
            k_msg_gemm<<<ntiles, 256, 0, stream>>>(x, Wt, H, N);
            k_scatter<<<(E + 7) / 8, 256, 0, stream>>>(H, src, dst, edge_attr, m, E, t);
        }
        k_gru<<<ntiles, 256, 0, stream>>>(m, x, W_ih, W_hh, b_ih, b_hh, N);
    }

    hipMemsetAsync(sums, 0, ((size_t)SEGS * CN + SEGS) * sizeof(float), stream);
    k_segsum<<<N, CN, 0, stream>>>(x, batch, x_type, sums, cnt, N);
    k_agg<<<SEGS, CN, 0, stream>>>(sums, cnt, w_attn, b_attn, wlog);
    k_head<<<BB, CN, 0, stream>>>(sums, wlog, W_enc, b_enc, W_dec, b_dec, out);
}